// CGAfterGatherConvolution_87351044866506
// MI455X (gfx1250) — compile-verified
//
#include <hip/hip_runtime.h>
#include <math.h>

typedef __attribute__((ext_vector_type(2))) float v2f;
typedef __attribute__((ext_vector_type(8))) float v8f;

#define N_NODES 5000
#define N_EDGES 80000
#define XD      288
#define XXW     2592      // 9*288 per node
#define OUTW    1632

// -------- shifted softplus: log(1+e^v) - log 2, numerically stable --------
__device__ __forceinline__ float ssp(float v) {
    return fmaxf(v, 0.0f) + log1pf(expf(-fabsf(v))) - 0.69314718055994531f;
}

// -------- f32 WMMA 16x16x4 (D = A(16x4) * B(4x16) + C) --------
__device__ __forceinline__ v8f wmma4(v2f a, v2f b, v8f c) {
    return __builtin_amdgcn_wmma_f32_16x16x4_f32(
        /*neg_a=*/false, a, /*neg_b=*/false, b,
        /*c_mod=*/(short)0, c, /*reuse_a=*/false, /*reuse_b=*/false);
}

// ---------------------------------------------------------------------------
// Kernel 0: zero the segment-sum accumulator xx (5000 x 9 x 288)
// ---------------------------------------------------------------------------
__global__ void zero_kernel(float* __restrict__ p, size_t n) {
    size_t i = (size_t)blockIdx.x * blockDim.x + threadIdx.x;
    size_t stride = (size_t)gridDim.x * blockDim.x;
    for (; i < n; i += stride) p[i] = 0.0f;
}

// ---------------------------------------------------------------------------
// Kernel 1: edge MLP.  One wave computes 16 edges:
//   h1 = ssp(emb@W1 / sqrt(8)); h2 = ssp(h1@W2 / 8); w = h2@W3 / 8
// A-operand lane layout (f32 16x4): lane L -> row (L&15), K pair base 2*(L>>4)
// B-operand: lane L -> col (L&15), same K pair base.
// D layout: vgpr r -> row r + 8*(L>=16), col (L&15).
// ---------------------------------------------------------------------------
__global__ __launch_bounds__(128)
void edge_mlp_kernel(const float* __restrict__ emb, const float* __restrict__ W1,
                     const float* __restrict__ W2, const float* __restrict__ W3,
                     float* __restrict__ wout)
{
    __shared__ float hbuf[4][16][68];   // 16x64 hidden per wave, padded rows
    const int wave = threadIdx.x >> 5;
    const int lane = threadIdx.x & 31;
    const int tile = blockIdx.x * 4 + wave;        // 16 edges per tile
    if (tile * 16 >= N_EDGES) return;              // wave-uniform exit
    const int lc = lane & 15;
    const int kb = (lane >> 4) << 1;               // 0 or 2
    const int hi = (lane >> 4) << 3;               // 0 or 8 (D row offset)
    float (*h)[68] = hbuf[wave];
    const float rs8  = 0.35355339059327373f;       // 1/sqrt(8)
    const float rs64 = 0.125f;                     // 1/sqrt(64)
    const int e_row = tile * 16 + lc;

    // ---- GEMM1: (16x8)@(8x64), K=8 -> 2 wmma steps per N tile ----
    #pragma unroll
    for (int nt = 0; nt < 4; ++nt) {
        v8f acc = {};
        #pragma unroll
        for (int kk = 0; kk < 2; ++kk) {
            const int k0 = kk * 4 + kb;
            v2f a, b;
            a.x = emb[e_row * 8 + k0];
            a.y = emb[e_row * 8 + k0 + 1];
            b.x = W1[k0 * 64 + nt * 16 + lc];
            b.y = W1[(k0 + 1) * 64 + nt * 16 + lc];
            acc = wmma4(a, b, acc);
        }
        #pragma unroll
        for (int r = 0; r < 8; ++r)
            h[r + hi][nt * 16 + lc] = ssp(acc[r] * rs8);
    }

    // ---- GEMM2: (16x64)@(64x64); read all of h, then overwrite in place ----
    v8f acc2[4];
    #pragma unroll
    for (int nt = 0; nt < 4; ++nt) {
        v8f acc = {};
        #pragma unroll
        for (int kk = 0; kk < 16; ++kk) {
            const int k0 = kk * 4 + kb;
            v2f a, b;
            a.x = h[lc][k0];
            a.y = h[lc][k0 + 1];
            b.x = W2[k0 * 64 + nt * 16 + lc];
            b.y = W2[(k0 + 1) * 64 + nt * 16 + lc];
            acc = wmma4(a, b, acc);
        }
        acc2[nt] = acc;
    }
    #pragma unroll
    for (int nt = 0; nt < 4; ++nt)
        #pragma unroll
        for (int r = 0; r < 8; ++r)
            h[r + hi][nt * 16 + lc] = ssp(acc2[nt][r] * rs64);

    // ---- GEMM3: (16x64)@(64x96) -> global w ----
    #pragma unroll
    for (int nt = 0; nt < 6; ++nt) {
        v8f acc = {};
        #pragma unroll
        for (int kk = 0; kk < 16; ++kk) {
            const int k0 = kk * 4 + kb;
            v2f a, b;
            a.x = h[lc][k0];
            a.y = h[lc][k0 + 1];
            b.x = W3[k0 * 96 + nt * 16 + lc];
            b.y = W3[(k0 + 1) * 96 + nt * 16 + lc];
            acc = wmma4(a, b, acc);
        }
        #pragma unroll
        for (int r = 0; r < 8; ++r)
            wout[(size_t)(tile * 16 + r + hi) * 96 + nt * 16 + lc] = acc[r] * rs64;
    }
}

// ---------------------------------------------------------------------------
// Kernel 2: per-edge kron + segment-sum scatter (atomics hit L2-resident xx)
// ---------------------------------------------------------------------------
__global__ __launch_bounds__(288)
void scatter_kernel(const float* __restrict__ x, const float* __restrict__ y,
                    const float* __restrict__ w, const int* __restrict__ esrc,
                    const int* __restrict__ edst, float* __restrict__ xx)
{
    const int e = blockIdx.x;
    const int c = threadIdx.x;                     // 0..287 channel
    const int src = esrc[e];
    const int dst = edst[e];
    // W_INDI: 32 x repeat1, 32 x repeat3, 32 x repeat5
    const int wi = (c < 32) ? c : (c < 128 ? 32 + (c - 32) / 3 : 64 + (c - 128) / 5);
    const float xv = x[(size_t)src * XD + c] * w[(size_t)e * 96 + wi];
    const float* ye = y + (size_t)e * 9;
    float* base = xx + (size_t)dst * XXW + c;
    #pragma unroll
    for (int i = 0; i < 9; ++i)
        atomicAdd(base + i * XD, ye[i] * xv);
}

// ---------------------------------------------------------------------------
// Kernel 3: per-node contraction as GEMM (160000 x K) @ (K x NOUT), WMMA f32,
// with the reshape/transpose folded into A addressing and the per-path output
// slicing + sqrt(d3) scaling folded into the D store.
// Final column for GEMM col 'col' of block b: BASE[b][col] + u*D3[b][col] + T[b][col]
// ---------------------------------------------------------------------------
__constant__ int C_D3[3][22] = {
    {1,3,3,3,5,5,5,5,5, 0,0,0,0,0,0,0,0,0,0,0,0,0},
    {3,3,3,1,3,3,3,5,5,5,5,5,3,3,3,5,5,5,5,5, 0,0},
    {5,5,5,5,5,3,3,3,5,5,5,5,5,1,3,3,3,5,5,5,5,5}};
__constant__ int C_T[3][22] = {
    {0,0,1,2,0,1,2,3,4, 0,0,0,0,0,0,0,0,0,0,0,0,0},
    {0,1,2,0,0,1,2,0,1,2,3,4,0,1,2,0,1,2,3,4, 0,0},
    {0,1,2,3,4,0,1,2,0,1,2,3,4,0,0,1,2,0,1,2,3,4}};
__constant__ int C_BASE[3][22] = {
    {0,32,32,32,128,128,128,128,128, 0,0,0,0,0,0,0,0,0,0,0,0,0},
    {288,288,288,384,416,416,416,512,512,512,512,512,672,672,672,768,768,768,768,768, 0,0},
    {928,928,928,928,928,1088,1088,1088,1184,1184,1184,1184,1184,1344,1376,1376,1376,1472,1472,1472,1472,1472}};

template<int BLK, int D1, int K, int NOUT, int NTILES, int START>
__global__ __launch_bounds__(128)
void contract_kernel(const float* __restrict__ xx, const float* __restrict__ cmat,
                     float* __restrict__ out)
{
    constexpr int KP = (K + 3) & ~3;               // K padded to multiple of 4
    const int wave = threadIdx.x >> 5;
    const int lane = threadIdx.x & 31;
    const int gw = blockIdx.x * 4 + wave;
    const int total = (N_NODES * 32 / 16) * NTILES;
    if (gw >= total) return;                       // wave-uniform exit
    const int mtile = gw / NTILES;
    const int nt    = gw % NTILES;
    const int lc = lane & 15;
    const int kb = (lane >> 4) << 1;
    const int hi = (lane >> 4) << 3;

    // A row for this lane: row = node*32 + u; A(row,k) = xx[z][k/D1][START + u*D1 + k%D1]
    const int arow = mtile * 16 + lc;
    const int z = arow >> 5, u = arow & 31;
    const float* xxz = xx + (size_t)z * XXW + START + u * D1;
    const int bcol = nt * 16 + lc;

    v8f acc = {};
    #pragma unroll
    for (int kk = 0; kk < KP / 4; ++kk) {
        const int k0 = kk * 4 + kb;
        const int k1 = k0 + 1;
        v2f a, b;
        a.x = (k0 < K) ? xxz[(k0 / D1) * XD + (k0 % D1)] : 0.0f;
        a.y = (k1 < K) ? xxz[(k1 / D1) * XD + (k1 % D1)] : 0.0f;
        b.x = (k0 < K && bcol < NOUT) ? cmat[k0 * NOUT + bcol] : 0.0f;
        b.y = (k1 < K && bcol < NOUT) ? cmat[k1 * NOUT + bcol] : 0.0f;
        acc = wmma4(a, b, acc);
    }

    if (bcol < NOUT) {
        const int d3   = C_D3[BLK][bcol];
        const int t    = C_T[BLK][bcol];
        const int base = C_BASE[BLK][bcol];
        const float pw = sqrtf((float)d3);
        #pragma unroll
        for (int r = 0; r < 8; ++r) {
            const int rg = mtile * 16 + r + hi;    // global row = node*32 + u
            const int zz = rg >> 5, uu = rg & 31;
            out[(size_t)zz * OUTW + base + uu * d3 + t] = acc[r] * pw;
        }
    }
}

// ---------------------------------------------------------------------------
extern "C" void kernel_launch(void* const* d_in, const int* in_sizes, int n_in,
                              void* d_out, int out_size, void* d_ws, size_t ws_size,
                              hipStream_t stream)
{
    const float* x    = (const float*)d_in[0];
    const float* y    = (const float*)d_in[1];
    const float* emb  = (const float*)d_in[2];
    const float* W1   = (const float*)d_in[3];
    const float* W2   = (const float*)d_in[4];
    const float* W3   = (const float*)d_in[5];
    const float* c0   = (const float*)d_in[6];
    const float* c1   = (const float*)d_in[7];
    const float* c2   = (const float*)d_in[8];
    const int*   esrc = (const int*)d_in[9];
    const int*   edst = (const int*)d_in[10];
    float* out = (float*)d_out;

    // workspace: w (80000x96 f32) then xx (5000x9x288 f32)
    float* w  = (float*)d_ws;
    float* xx = w + (size_t)N_EDGES * 96;

    zero_kernel<<<4096, 256, 0, stream>>>(xx, (size_t)N_NODES * XXW);

    edge_mlp_kernel<<<(N_EDGES / 16 + 3) / 4, 128, 0, stream>>>(emb, W1, W2, W3, w);

    scatter_kernel<<<N_EDGES, 288, 0, stream>>>(x, y, w, esrc, edst, xx);

    const int mt = N_NODES * 32 / 16;  // 10000 row tiles
    contract_kernel<0, 1,  9,  9, 1,   0><<<(mt * 1 + 3) / 4, 128, 0, stream>>>(xx, c0, out);
    contract_kernel<1, 3, 27, 20, 2,  32><<<(mt * 2 + 3) / 4, 128, 0, stream>>>(xx, c1, out);
    contract_kernel<2, 5, 45, 22, 2, 128><<<(mt * 2 + 3) / 4, 128, 0, stream>>>(xx, c2, out);
}